// S4ConditionalSideChainModel_2413771620884
// MI455X (gfx1250) — compile-verified
//
#include <hip/hip_runtime.h>

// ---------------------------------------------------------------------------
// S4 conditional side-chain model for MI455X (gfx1250), compile-only target.
//
// Strategy:
//  * FFT conv -> exact diagonal SSM recurrence, parallelized as a 3-pass
//    chunked scan (chunk=256, 32768 parallel chunk tasks per layer).
//  * lin GEMM (262144x32x32 fp32) on V_WMMA_F32_16X16X4_F32, fused
//    bias+gelu, storing transposed (B,H,L) for the scan.
//  * Branch-free gelu: x * sigmoid(1.59576912*(x+0.044715 x^3)) using
//    v_exp_f32 + v_rcp_f32 (no EXEC divergence in hot loops).
//  * Everything stays L2-resident (~72 MB workspace << 192 MB L2).
// ---------------------------------------------------------------------------

constexpr int kB    = 8;
constexpr int kL    = 32768;
constexpr int kH    = 32;
constexpr int kN    = 16;
constexpr int kDep  = 4;
constexpr int kHid  = 32;
constexpr int kMlpD = 2;
constexpr int kCH   = 256;          // scan chunk length
constexpr int kNC   = kL / kCH;     // 128 chunks per sequence
constexpr int kSeq  = kB * kH;      // 256 independent sequences

typedef __attribute__((ext_vector_type(2))) float v2f;
typedef __attribute__((ext_vector_type(8))) float v8f;

// tanh-gelu == x * sigmoid(2*sqrt(2/pi)*(x + 0.044715 x^3)); branch-free via
// hardware exp2/rcp (v_exp_f32, v_rcp_f32). Tails saturate correctly.
__device__ __forceinline__ float gelu_fast(float x) {
  float z = 1.5957691216057308f * (x + 0.044715f * x * x * x);
  float e = __builtin_amdgcn_exp2f(z * -1.4426950408889634f);  // exp(-z)
  return x * __builtin_amdgcn_rcpf(1.0f + e);                  // x * sigmoid(z)
}

// ---------------------------------------------------------------------------
// Conditioning MLP (B=8, 2->32->32->32, gelu) + FiLM gamma/beta for all layers.
// One block of 256 threads; thread (b, j).
// ---------------------------------------------------------------------------
__global__ void cond_film_kernel(const float* __restrict__ cp,
                                 const float* __restrict__ Win,
                                 const float* __restrict__ bin,
                                 const float* __restrict__ Whid,
                                 const float* __restrict__ bhid,
                                 const float* __restrict__ filmW,
                                 const float* __restrict__ filmb,
                                 float* __restrict__ gbbuf) {
  __shared__ float cond[kB][kHid];
  int t = threadIdx.x;
  int b = t >> 5, j = t & 31;
  float v = cp[b * 2 + 0] * Win[j * 2 + 0] + cp[b * 2 + 1] * Win[j * 2 + 1] + bin[j];
  cond[b][j] = gelu_fast(v);
  __syncthreads();
  for (int d = 0; d < kMlpD; ++d) {
    float acc = bhid[d * kHid + j];
    for (int k = 0; k < kHid; ++k) acc += cond[b][k] * Whid[(d * kHid + j) * kHid + k];
    __syncthreads();
    cond[b][j] = gelu_fast(acc);
    __syncthreads();
  }
  // FiLM: gb[d,b,jj] = film_b + cond[b,:] . film_W[d,jj,:]
  for (int d = 0; d < kDep; ++d) {
    for (int half = 0; half < 2; ++half) {
      int jj = half * kH + j;
      float acc = filmb[d * 2 * kH + jj];
      for (int k = 0; k < kHid; ++k) acc += cond[b][k] * filmW[(d * 2 * kH + jj) * kHid + k];
      gbbuf[(d * kB + b) * (2 * kH) + jj] = acc;
    }
  }
}

// ---------------------------------------------------------------------------
// Per (layer,h,n): w = exp(dt*A), coef = C*(w-1)/A, wp = w^kCH (8 squarings).
// ---------------------------------------------------------------------------
__global__ void ssm_params_kernel(const float* __restrict__ log_dt,
                                  const float* __restrict__ Are,
                                  const float* __restrict__ Aim,
                                  const float* __restrict__ Cre,
                                  const float* __restrict__ Cim,
                                  float* __restrict__ wbuf,
                                  float* __restrict__ cbuf,
                                  float* __restrict__ wpbuf) {
  int g = blockIdx.x * blockDim.x + threadIdx.x;
  if (g >= kDep * kH * kN) return;
  int dh = g / kN;
  float dt = expf(log_dt[dh]);
  float ar = Are[g], ai = Aim[g];
  float xr = dt * ar, xi = dt * ai;
  float e = expf(xr);
  float wr = e * cosf(xi), wi = e * sinf(xi);
  float cr0 = Cre[g], ci0 = Cim[g];
  float tr = cr0 * (wr - 1.0f) - ci0 * wi;
  float ti = cr0 * wi + ci0 * (wr - 1.0f);
  float den = ar * ar + ai * ai;
  float cr = (tr * ar + ti * ai) / den;
  float ci = (ti * ar - tr * ai) / den;
  float pr = wr, pi = wi;
  #pragma unroll
  for (int k = 0; k < 8; ++k) { // w^(2^8) = w^256 = w^kCH
    float nr = pr * pr - pi * pi;
    float ni = 2.0f * pr * pi;
    pr = nr; pi = ni;
  }
  wbuf[2 * g] = wr;  wbuf[2 * g + 1] = wi;
  cbuf[2 * g] = cr;  cbuf[2 * g + 1] = ci;
  wpbuf[2 * g] = pr; wpbuf[2 * g + 1] = pi;
}

// ---------------------------------------------------------------------------
// h0[b,l,h] = x[b,l]*exp_W[h] + exp_b[h]   (fully coalesced, 1 elem/thread)
// ---------------------------------------------------------------------------
__global__ void init_h_kernel(const float* __restrict__ x,
                              const float* __restrict__ expW,
                              const float* __restrict__ expb,
                              float* __restrict__ hbuf) {
  int e = blockIdx.x * blockDim.x + threadIdx.x;
  if (e >= kB * kL * kH) return;
  int h = e & 31;
  int bl = e >> 5;
  hbuf[e] = x[bl] * expW[h] + expb[h];
}

// ---------------------------------------------------------------------------
// u = gelu(h @ W^T + bias) via V_WMMA_F32_16X16X4_F32, stored as (B,H,L).
// One wave per 16-row M-tile; 8 K-steps x 2 N-tiles = 16 WMMAs/wave.
// A 16x4 f32 layout: lanes0-15 row=lane K={0,1}; lanes16-31 row=lane-16 K={2,3}.
// C/D layout: VGPR j -> (M=j, N=lane) / (M=j+8, N=lane-16)  => each lane's 8
// accumulator values are 8 consecutive l's at fixed column n: float4 stores.
// ---------------------------------------------------------------------------
__global__ void lin_gemm_kernel(const float* __restrict__ hbuf,
                                const float* __restrict__ W,     // lin_W[d] (32x32)
                                const float* __restrict__ bias,  // lin_b[d]
                                float* __restrict__ ubuf) {
  int lane = threadIdx.x & 31;
  int wid  = threadIdx.x >> 5;
  int tile = blockIdx.x * 8 + wid;       // 16384 tiles of 16 rows
  int row0 = tile * 16;
  int b  = row0 / kL;
  int l0 = row0 % kL;                    // L%16==0 -> tiles never cross batch
  int mrow = lane & 15;
  int ksel = lane >> 4;
  int n0   = lane & 15;

  const float* arow = hbuf + (size_t)(row0 + mrow) * kH;
  v8f acc0 = {0.f, 0.f, 0.f, 0.f, 0.f, 0.f, 0.f, 0.f};
  v8f acc1 = {0.f, 0.f, 0.f, 0.f, 0.f, 0.f, 0.f, 0.f};
  #pragma unroll
  for (int ks = 0; ks < 8; ++ks) {
    int kb = ks * 4 + ksel * 2;
    float2 af  = *reinterpret_cast<const float2*>(arow + kb);
    float2 b0f = *reinterpret_cast<const float2*>(W + n0 * kH + kb);
    float2 b1f = *reinterpret_cast<const float2*>(W + (n0 + 16) * kH + kb);
    v2f a;  a[0] = af.x;  a[1] = af.y;
    v2f b0; b0[0] = b0f.x; b0[1] = b0f.y;
    v2f b1; b1[0] = b1f.x; b1[1] = b1f.y;
    acc0 = __builtin_amdgcn_wmma_f32_16x16x4_f32(false, a, false, b0, (short)0, acc0, false, false);
    acc1 = __builtin_amdgcn_wmma_f32_16x16x4_f32(false, a, false, b1, (short)0, acc1, false, false);
  }
  float bias0 = bias[n0];
  float bias1 = bias[n0 + 16];
  int lbase = l0 + ksel * 8;
  float* u0 = ubuf + (size_t)(b * kH + n0) * kL + lbase;
  float* u1 = ubuf + (size_t)(b * kH + n0 + 16) * kL + lbase;
  float4 o;
  o.x = gelu_fast(acc0[0] + bias0); o.y = gelu_fast(acc0[1] + bias0);
  o.z = gelu_fast(acc0[2] + bias0); o.w = gelu_fast(acc0[3] + bias0);
  *reinterpret_cast<float4*>(u0) = o;
  o.x = gelu_fast(acc0[4] + bias0); o.y = gelu_fast(acc0[5] + bias0);
  o.z = gelu_fast(acc0[6] + bias0); o.w = gelu_fast(acc0[7] + bias0);
  *reinterpret_cast<float4*>(u0 + 4) = o;
  o.x = gelu_fast(acc1[0] + bias1); o.y = gelu_fast(acc1[1] + bias1);
  o.z = gelu_fast(acc1[2] + bias1); o.w = gelu_fast(acc1[3] + bias1);
  *reinterpret_cast<float4*>(u1) = o;
  o.x = gelu_fast(acc1[4] + bias1); o.y = gelu_fast(acc1[5] + bias1);
  o.z = gelu_fast(acc1[6] + bias1); o.w = gelu_fast(acc1[7] + bias1);
  *reinterpret_cast<float4*>(u1 + 4) = o;
}

// ---------------------------------------------------------------------------
// Scan pass 1: per chunk, local final state (zero init). Thread = (b,h,chunk),
// lane = head so all wave lanes walk the same chunk index.
// ---------------------------------------------------------------------------
__global__ void ssm_pass1_kernel(const float* __restrict__ ubuf,
                                 const float* __restrict__ wbuf,   // +layer
                                 float* __restrict__ states) {
  int g = blockIdx.x * blockDim.x + threadIdx.x;  // kSeq*kNC = 32768
  int h = g & 31;
  int b = (g >> 5) & 7;
  int c = g >> 8;
  int seq = b * kH + h;
  const float4* up = reinterpret_cast<const float4*>(ubuf + (size_t)seq * kL + (size_t)c * kCH);
  const float* wp = wbuf + h * kN * 2;
  float wr[kN], wi[kN], sr[kN], si[kN];
  #pragma unroll
  for (int n = 0; n < kN; ++n) { wr[n] = wp[2 * n]; wi[n] = wp[2 * n + 1]; sr[n] = 0.f; si[n] = 0.f; }
  for (int i = 0; i < kCH / 4; ++i) {
    float4 uv = up[i];
    float us[4] = {uv.x, uv.y, uv.z, uv.w};
    #pragma unroll
    for (int t = 0; t < 4; ++t) {
      float uval = us[t];
      #pragma unroll
      for (int n = 0; n < kN; ++n) {
        float nr = wr[n] * sr[n] - wi[n] * si[n] + uval;
        float ni = wr[n] * si[n] + wi[n] * sr[n];
        sr[n] = nr; si[n] = ni;
      }
    }
  }
  float* st = states + ((size_t)seq * kNC + c) * kN * 2;
  #pragma unroll
  for (int n = 0; n < kN; ++n) { st[2 * n] = sr[n]; st[2 * n + 1] = si[n]; }
}

// ---------------------------------------------------------------------------
// Scan pass 2: inter-chunk combine, in place: states[c] := incoming state of
// chunk c; carry' = w^kCH * carry + local. Thread = (seq, n), 4096 threads.
// ---------------------------------------------------------------------------
__global__ void ssm_pass2_kernel(const float* __restrict__ wpbuf,  // +layer
                                 float* __restrict__ states) {
  int g = blockIdx.x * blockDim.x + threadIdx.x;  // kSeq*kN = 4096
  int n = g & 15;
  int seq = g >> 4;
  int h = seq & 31;
  float pr = wpbuf[(h * kN + n) * 2], pi = wpbuf[(h * kN + n) * 2 + 1];
  float cr = 0.f, ci = 0.f;
  for (int c = 0; c < kNC; ++c) {
    size_t idx = ((size_t)seq * kNC + c) * kN * 2 + n * 2;
    float lr = states[idx], li = states[idx + 1];
    states[idx] = cr; states[idx + 1] = ci;
    float nr = pr * cr - pi * ci + lr;
    float ni = pr * ci + pi * cr + li;
    cr = nr; ci = ni;
  }
}

// ---------------------------------------------------------------------------
// Scan pass 3: recompute with true initial state; y = 2*Re(coef.s) + D*u;
// FiLM + gelu; write h[b,l,:] (lane=head -> 128B coalesced row per step).
// ---------------------------------------------------------------------------
__global__ void ssm_pass3_kernel(const float* __restrict__ ubuf,
                                 const float* __restrict__ wbuf,   // +layer
                                 const float* __restrict__ cbuf,   // +layer
                                 const float* __restrict__ Dv,     // s4_D[d]
                                 const float* __restrict__ gb,     // gbbuf[d]
                                 const float* __restrict__ states,
                                 float* __restrict__ hbuf) {
  int g = blockIdx.x * blockDim.x + threadIdx.x;
  int h = g & 31;
  int b = (g >> 5) & 7;
  int c = g >> 8;
  int seq = b * kH + h;
  const float* wp = wbuf + h * kN * 2;
  const float* cp = cbuf + h * kN * 2;
  const float* st = states + ((size_t)seq * kNC + c) * kN * 2;
  float wr[kN], wi[kN], cr[kN], ci[kN], sr[kN], si[kN];
  #pragma unroll
  for (int n = 0; n < kN; ++n) {
    wr[n] = wp[2 * n]; wi[n] = wp[2 * n + 1];
    cr[n] = cp[2 * n]; ci[n] = cp[2 * n + 1];
    sr[n] = st[2 * n]; si[n] = st[2 * n + 1];
  }
  float Dh    = Dv[h];
  float gamma = gb[b * (2 * kH) + h];
  float beta  = gb[b * (2 * kH) + kH + h];
  const float4* up = reinterpret_cast<const float4*>(ubuf + (size_t)seq * kL + (size_t)c * kCH);
  float* out = hbuf + ((size_t)(b * kL + c * kCH)) * kH + h;
  for (int i = 0; i < kCH / 4; ++i) {
    float4 uv = up[i];
    float us[4] = {uv.x, uv.y, uv.z, uv.w};
    #pragma unroll
    for (int t = 0; t < 4; ++t) {
      float uval = us[t];
      float y = 0.f;
      #pragma unroll
      for (int n = 0; n < kN; ++n) {
        float nr = wr[n] * sr[n] - wi[n] * si[n] + uval;
        float ni = wr[n] * si[n] + wi[n] * sr[n];
        sr[n] = nr; si[n] = ni;
        y += cr[n] * nr - ci[n] * ni;
      }
      y = 2.0f * y + Dh * uval;
      out[(size_t)(i * 4 + t) * kH] = gelu_fast(gamma * y + beta);
    }
  }
}

// ---------------------------------------------------------------------------
// out[b,l] = h[b,l,:] . con_W + con_b
// ---------------------------------------------------------------------------
__global__ void out_kernel(const float* __restrict__ hbuf,
                           const float* __restrict__ conW,
                           const float* __restrict__ conb,
                           float* __restrict__ out) {
  int g = blockIdx.x * blockDim.x + threadIdx.x;
  if (g >= kB * kL) return;
  const float4* hp = reinterpret_cast<const float4*>(hbuf + (size_t)g * kH);
  const float4* wp = reinterpret_cast<const float4*>(conW);
  float acc = conb[0];
  #pragma unroll
  for (int q = 0; q < 8; ++q) {
    float4 hv = hp[q];
    float4 wv = wp[q];
    acc += hv.x * wv.x + hv.y * wv.y + hv.z * wv.z + hv.w * wv.w;
  }
  out[g] = acc;
}

extern "C" void kernel_launch(void* const* d_in, const int* in_sizes, int n_in,
                              void* d_out, int out_size, void* d_ws, size_t ws_size,
                              hipStream_t stream) {
  (void)in_sizes; (void)n_in; (void)out_size; (void)ws_size;
  const float* x        = (const float*)d_in[0];
  const float* cp       = (const float*)d_in[1];
  const float* mlp_W_in = (const float*)d_in[2];
  const float* mlp_b_in = (const float*)d_in[3];
  const float* mlp_W_hid= (const float*)d_in[4];
  const float* mlp_b_hid= (const float*)d_in[5];
  const float* exp_W    = (const float*)d_in[6];
  const float* exp_b    = (const float*)d_in[7];
  const float* con_W    = (const float*)d_in[8];
  const float* con_b    = (const float*)d_in[9];
  const float* lin_W    = (const float*)d_in[10];
  const float* lin_b    = (const float*)d_in[11];
  const float* log_dt   = (const float*)d_in[12];
  const float* A_re     = (const float*)d_in[13];
  const float* A_im     = (const float*)d_in[14];
  const float* C_re     = (const float*)d_in[15];
  const float* C_im     = (const float*)d_in[16];
  const float* s4_D     = (const float*)d_in[17];
  const float* film_W   = (const float*)d_in[18];
  const float* film_b   = (const float*)d_in[19];
  float* out = (float*)d_out;

  // Workspace carve-up (~72 MB, fits L2-resident working set).
  float* ws = (float*)d_ws;
  size_t off = 0;
  float* hbuf   = ws + off; off += (size_t)kB * kL * kH;          // 33.5 MB
  float* ubuf   = ws + off; off += (size_t)kB * kL * kH;          // 33.5 MB
  float* states = ws + off; off += (size_t)kSeq * kNC * kN * 2;   // 4 MB
  float* wbuf   = ws + off; off += (size_t)kDep * kH * kN * 2;
  float* cbuf   = ws + off; off += (size_t)kDep * kH * kN * 2;
  float* wpbuf  = ws + off; off += (size_t)kDep * kH * kN * 2;
  float* gbbuf  = ws + off; off += (size_t)kDep * kB * 2 * kH;

  cond_film_kernel<<<1, 256, 0, stream>>>(cp, mlp_W_in, mlp_b_in, mlp_W_hid,
                                          mlp_b_hid, film_W, film_b, gbbuf);
  ssm_params_kernel<<<(kDep * kH * kN + 255) / 256, 256, 0, stream>>>(
      log_dt, A_re, A_im, C_re, C_im, wbuf, cbuf, wpbuf);
  init_h_kernel<<<(kB * kL * kH + 255) / 256, 256, 0, stream>>>(x, exp_W, exp_b, hbuf);

  for (int d = 0; d < kDep; ++d) {
    lin_gemm_kernel<<<(kB * kL / 16) / 8, 256, 0, stream>>>(
        hbuf, lin_W + d * kH * kH, lin_b + d * kH, ubuf);
    ssm_pass1_kernel<<<(kSeq * kNC) / 256, 256, 0, stream>>>(
        ubuf, wbuf + d * kH * kN * 2, states);
    ssm_pass2_kernel<<<(kSeq * kN) / 256, 256, 0, stream>>>(
        wpbuf + d * kH * kN * 2, states);
    ssm_pass3_kernel<<<(kSeq * kNC) / 256, 256, 0, stream>>>(
        ubuf, wbuf + d * kH * kN * 2, cbuf + d * kH * kN * 2,
        s4_D + d * kH, gbbuf + d * kB * 2 * kH, states, hbuf);
  }
  out_kernel<<<(kB * kL + 255) / 256, 256, 0, stream>>>(hbuf, con_W, con_b, out);
}